// BatchSquareDiagonal_63032940036672
// MI455X (gfx1250) — compile-verified
//
#include <hip/hip_runtime.h>
#include <hip/hip_bf16.h>

// out[b] = sum_n d[b,n] * x[b,n]^2     (B=16384, N=2048, fp32)
//
// Memory-bound: 268 MB streamed once -> ~11.5 us floor at 23.3 TB/s.
// - float4 (global_load_b128) coalesced non-temporal loads (wave32 x 16B = 512B/issue)
// - scalar-counted hot loop (uniform trip count -> s_cbranch, no exec masking)
// - per-lane v_fma_f32 accumulation
// - cross-lane reduction via V_WMMA_F32_16X16X4_F32 with ones B-matrix
// - halves combined with ds_swizzle SWAPX16, lane 0 stores

typedef __attribute__((ext_vector_type(4))) float v4f;
typedef __attribute__((ext_vector_type(2))) float v2f;
typedef __attribute__((ext_vector_type(8))) float v8f;

__global__ __launch_bounds__(256) void
batch_sq_diag_kernel(const float* __restrict__ vec,
                     const float* __restrict__ diag,
                     float* __restrict__ out,
                     int N, int B) {
    const int lane = threadIdx.x & 31;
    const int wave = threadIdx.x >> 5;
    const int row  = blockIdx.x * 8 + wave;   // 8 wave32 per block, 1 row per wave
    if (row >= B) return;                     // wave-uniform -> EXEC stays all-1s

    const v4f* __restrict__ px = (const v4f*)(vec  + (size_t)row * N) + lane;
    const v4f* __restrict__ pd = (const v4f*)(diag + (size_t)row * N) + lane;
    const int nvec  = N >> 2;                 // float4 per row (512)
    const int iters = nvec >> 5;              // wave-uniform trip count (16)

    float acc = 0.0f;
    // Uniform trip count: loop control is pure SALU (s_cmp + s_cbranch), no
    // per-iteration exec manipulation. NT hint: data is streamed exactly once
    // (268 MB > 192 MB L2) -> don't pollute the cache hierarchy.
    #pragma unroll 4
    for (int j = 0; j < iters; ++j) {
        v4f x = __builtin_nontemporal_load(px + (size_t)j * 32);
        v4f d = __builtin_nontemporal_load(pd + (size_t)j * 32);
        acc = fmaf(d.x * x.x, x.x, acc);
        acc = fmaf(d.y * x.y, x.y, acc);
        acc = fmaf(d.z * x.z, x.z, acc);
        acc = fmaf(d.w * x.w, x.w, acc);
    }
    // Generic tail (empty for N=2048)
    for (int i = (iters << 5) + lane; i < nvec; i += 32) {
        v4f x = __builtin_nontemporal_load(px - lane + i);
        v4f d = __builtin_nontemporal_load(pd - lane + i);
        acc = fmaf(d.x * x.x, x.x, acc);
        acc = fmaf(d.y * x.y, x.y, acc);
        acc = fmaf(d.z * x.z, x.z, acc);
        acc = fmaf(d.w * x.w, x.w, acc);
    }

    // ---- Cross-lane reduction via WMMA ----
    // A (16x4 f32, 2 VGPRs/lane): VGPR0 lanes0-15 -> A[m][0]=acc(m),
    // lanes16-31 -> A[m][2]=acc(m+16); VGPR1 = 0 (K=1,3).
    // B = all ones (constant => B lane layout irrelevant).
    // => D[m][n] = acc(m) + acc(m+16) for every n.
    v2f a; a.x = acc;  a.y = 0.0f;
    v2f b; b.x = 1.0f; b.y = 1.0f;
    v8f c = {};
    c = __builtin_amdgcn_wmma_f32_16x16x4_f32(
            /*neg_a=*/false, a, /*neg_b=*/false, b,
            /*c_mod=*/(short)0, c, /*reuse_a=*/false, /*reuse_b=*/false);

    // D VGPR j: lanes0-15 hold M=j, lanes16-31 hold M=8+j. Per-lane sum of the
    // 8 VGPRs: lanes0-15 = sum acc(0..7,16..23); lanes16-31 = the rest.
    float s = c[0] + c[1] + c[2] + c[3] + c[4] + c[5] + c[6] + c[7];

    // Combine halves with ds_swizzle SWAPX16 (offset: xor=0x10, and=0x1f).
    int si = __builtin_amdgcn_ds_swizzle(__float_as_int(s), 0x401F);
    s += __int_as_float(si);

    if (lane == 0) out[row] = s;
}

extern "C" void kernel_launch(void* const* d_in, const int* in_sizes, int n_in,
                              void* d_out, int out_size, void* d_ws, size_t ws_size,
                              hipStream_t stream) {
    const float* vec  = (const float*)d_in[0];   // [B, N] fp32
    const float* diag = (const float*)d_in[1];   // [B, N] fp32
    float* out = (float*)d_out;                  // [B, 1] fp32

    const int B = out_size;                      // 16384
    const int N = in_sizes[0] / B;               // 2048

    dim3 block(256);                             // 8 wave32 -> 8 rows per block
    dim3 grid((B + 7) / 8);                      // 2048 blocks
    batch_sq_diag_kernel<<<grid, block, 0, stream>>>(vec, diag, out, N, B);
}